// CosineWeights_17901423690446
// MI455X (gfx1250) — compile-verified
//
#include <hip/hip_runtime.h>
#include <math.h>

typedef float v2f __attribute__((ext_vector_type(2)));
typedef float v8f __attribute__((ext_vector_type(8)));

#define BB   32
#define HH   8
#define JJ   8192
#define KK   128
#define EPSF 1e-6f
#define WAVES 8
#define JTILE 16
#define JBLK  (WAVES * JTILE)   // 128 j's per block

// ---------------------------------------------------------------------------
// Kernel 1: per (batch, 128-j chunk) block. Each of the 8 waves owns one
// 16-j tile and runs two fp32 WMMA accumulation chains over K=128:
//   C1[j,h] = sum_k mem[j,k]   * (mask^2*keys)[h,k]
//   C2[j,h] = sum_k mem[j,k]^2 * (mask^2)[h,k]
// then writes sharp = proj/(kn*sqrt(C2)+eps) * softplus(strength) to out.
// ---------------------------------------------------------------------------
__global__ __launch_bounds__(256) void cw_score_kernel(
    const float* __restrict__ memory,
    const float* __restrict__ keys,
    const float* __restrict__ strengths,
    const float* __restrict__ mask,
    float* __restrict__ sharp_out)
{
    __shared__ float wlds [16 * KK];   // rows 0..7: mask^2*keys ; rows 8..15: 0
    __shared__ float m2lds[16 * KK];   // rows 0..7: mask^2      ; rows 8..15: 0
    __shared__ float knorm[16];
    __shared__ float spl  [16];

    const int i   = blockIdx.y;
    const int jc  = blockIdx.x;
    const int tid = threadIdx.x;

    // --- preamble: stage B-operands (zero-padded to 16 heads) in LDS ---
    for (int e = tid; e < 16 * KK; e += 256) {
        const int h = e >> 7;          // / 128
        const int k = e & (KK - 1);
        float w = 0.f, m2 = 0.f;
        if (h < HH) {
            const float mk = mask[(i * HH + h) * KK + k];
            const float ky = keys[(i * HH + h) * KK + k];
            m2 = mk * mk;
            w  = m2 * ky;
        }
        wlds [e] = w;
        m2lds[e] = m2;
    }
    if (tid < 16) {
        float kn = 0.f, sp = 0.f;
        if (tid < HH) {
            const int base = (i * HH + tid) * KK;
            float s = 0.f;
            for (int k = 0; k < KK; ++k) {
                const float t = mask[base + k] * keys[base + k];
                s += t * t;
            }
            kn = sqrtf(s);                               // ||mask*keys||
            const float st = strengths[i * HH + tid];
            sp = (st > 20.f) ? st : log1pf(__expf(st));  // softplus
        }
        knorm[tid] = kn;
        spl  [tid] = sp;
    }
    __syncthreads();

    const int wave  = tid >> 5;
    const int lane  = tid & 31;
    const int j0    = jc * JBLK + wave * JTILE;
    const int row   = j0 + (lane & 15);     // A-matrix M index (ISA 16x4 f32 layout)
    const int khalf = (lane >> 4) << 1;     // lanes 0-15 -> K0,K1 ; 16-31 -> K2,K3
    const int n     = lane & 15;            // B/C N index

    const float* aptr = memory + (size_t)(i * JJ + row) * KK + khalf;
    const float* b1p  = &wlds [n * KK + khalf];
    const float* b2p  = &m2lds[n * KK + khalf];

    v8f c1 = {};
    v8f c2 = {};
#pragma unroll
    for (int s = 0; s < KK / 4; ++s) {
        v2f a  = *(const v2f*)(aptr + s * 4);
        v2f b1 = *(const v2f*)(b1p  + s * 4);
        v2f b2 = *(const v2f*)(b2p  + s * 4);
        v2f a2 = a * a;
        c1 = __builtin_amdgcn_wmma_f32_16x16x4_f32(false, a,  false, b1,
                                                   (short)0, c1, false, false);
        c2 = __builtin_amdgcn_wmma_f32_16x16x4_f32(false, a2, false, b2,
                                                   (short)0, c2, false, false);
    }

    // C layout: VGPR r -> M = r (lanes 0-15) / r+8 (lanes 16-31), N = lane%16
    if (n < HH) {
        const float kn    = knorm[n];
        const float sp    = spl[n];
        const int   jbase = j0 + ((lane >> 4) << 3);
        float* outp = sharp_out + (size_t)(i * HH + n) * JJ;
#pragma unroll
        for (int r = 0; r < 8; ++r) {
            const float proj = c1[r];
            const float mn   = sqrtf(c2[r]);
            const float sim  = proj / (kn * mn + EPSF);
            outp[jbase + r]  = sim * sp;
        }
    }
}

// ---------------------------------------------------------------------------
// Kernel 2: deterministic in-place softmax over J=8192 per (i,h) row.
// One block per row; values held in registers; LDS tree reductions.
// ---------------------------------------------------------------------------
__global__ __launch_bounds__(256) void cw_softmax_kernel(float* __restrict__ data)
{
    __shared__ float red[256];
    const int tid = threadIdx.x;
    float* p = data + (size_t)blockIdx.x * JJ;

    float v[JJ / 256];
    float mx = -INFINITY;
#pragma unroll
    for (int t = 0; t < JJ / 256; ++t) {
        v[t] = p[tid + t * 256];
        mx = fmaxf(mx, v[t]);
    }
    red[tid] = mx;
    __syncthreads();
    for (int s = 128; s > 0; s >>= 1) {
        if (tid < s) red[tid] = fmaxf(red[tid], red[tid + s]);
        __syncthreads();
    }
    mx = red[0];
    __syncthreads();

    float sum = 0.f;
#pragma unroll
    for (int t = 0; t < JJ / 256; ++t) {
        v[t] = __expf(v[t] - mx);
        sum += v[t];
    }
    red[tid] = sum;
    __syncthreads();
    for (int s = 128; s > 0; s >>= 1) {
        if (tid < s) red[tid] += red[tid + s];
        __syncthreads();
    }
    const float inv = 1.f / red[0];
#pragma unroll
    for (int t = 0; t < JJ / 256; ++t) {
        p[tid + t * 256] = v[t] * inv;
    }
}

extern "C" void kernel_launch(void* const* d_in, const int* in_sizes, int n_in,
                              void* d_out, int out_size, void* d_ws, size_t ws_size,
                              hipStream_t stream)
{
    const float* memory    = (const float*)d_in[0];
    const float* keys      = (const float*)d_in[1];
    const float* strengths = (const float*)d_in[2];
    const float* mask      = (const float*)d_in[3];
    float* out = (float*)d_out;

    dim3 grid1(JJ / JBLK, BB);                 // 64 x 32 blocks
    cw_score_kernel<<<grid1, 256, 0, stream>>>(memory, keys, strengths, mask, out);
    cw_softmax_kernel<<<BB * HH, 256, 0, stream>>>(out);
}